// ImplicitModalFilter_86706799772315
// MI455X (gfx1250) — compile-verified
//
#include <hip/hip_runtime.h>

// ImplicitModalFilter on MI455X (gfx1250), WMMA f32 16x16x4 formulation.
//
// out[d, l] = sum_o R[d,o] * exp(glogp[d,o] * l),  glogp = -exp(p+gamma)
// Decompose l = 16*m + n + 256*j  (m,n,j in [0,16)):
//   A[m, o]   = exp(glogp[o] * 16m)                 (constant per d)
//   B_j[o, n] = R[o] * exp(glogp[o] * (n + 256j))   (B_{j+1} = B_j * exp(glogp*256))
//   C = A x B_j  via 16 chained V_WMMA_F32_16X16X4_F32 (K covers o=0..63)
// One wave handles one d-row; 16 j-tiles of 256 outputs each.

typedef __attribute__((ext_vector_type(2))) float v2f;
typedef __attribute__((ext_vector_type(8))) float v8f;

#define D_MODEL 2048
#define ORDER   64
#define SEQ_L   4096
#define WAVES_PER_BLOCK 8

__global__ __launch_bounds__(WAVES_PER_BLOCK * 32)
void ImplicitModalFilter_wmma_kernel(const float* __restrict__ gamma,
                                     const float* __restrict__ R,
                                     const float* __restrict__ p,
                                     float* __restrict__ out)
{
    const int lane = threadIdx.x & 31;
    const int wave = threadIdx.x >> 5;
    const int d    = blockIdx.x * WAVES_PER_BLOCK + wave;
    const int h    = lane >> 4;   // which 16-lane half
    const int m    = lane & 15;   // A row index / B,C column index

    const float* grow = gamma + (size_t)d * ORDER;
    const float* Rrow = R     + (size_t)d * ORDER;
    const float* prow = p     + (size_t)d * ORDER;

    // Per-lane WMMA operand state.
    // A 16x4 f32 layout: lane m = M, VGPR r + half h -> K = r + 2h  => o = 4*kk + 2h + r
    // B  4x16 f32 layout (assumed symmetric): lane n = N, K = r + 2h => same o formula
    v2f a[16];      // A[m, o]            (fixed)
    v2f b[16];      // B_j[o, n]          (rescaled each j)
    v2f d256[16];   // exp(glogp[o]*256)  (fixed)

#pragma unroll
    for (int kk = 0; kk < 16; ++kk) {
#pragma unroll
        for (int r = 0; r < 2; ++r) {
            const int o = 4 * kk + 2 * h + r;
            const float gv = grow[o];
            const float pv = prow[o];
            const float Rv = Rrow[o];
            const float glogp = -__expf(pv + gv);          // -exp(p)*exp(gamma)
            a[kk][r]    = __expf(glogp * (16.0f * (float)m));
            b[kk][r]    = Rv * __expf(glogp * (float)m);   // n == lane&15 == m
            d256[kk][r] = __expf(glogp * 256.0f);
        }
    }

    // C/D 16x16 f32 layout: VGPR r, half h -> M = r + 8h, N = lane&15.
    // l = 16*M + N + 256*j
    float* orow = out + (size_t)d * SEQ_L + 128 * h + m;

    for (int j = 0; j < 16; ++j) {
        v8f c = {};
#pragma unroll
        for (int kk = 0; kk < 16; ++kk) {
            // 8 args: (neg_a, A, neg_b, B, c_mod, C, reuse_a, reuse_b)
            c = __builtin_amdgcn_wmma_f32_16x16x4_f32(
                    false, a[kk], false, b[kk], (short)0, c, false, false);
        }
#pragma unroll
        for (int r = 0; r < 8; ++r) {
            orow[j * 256 + r * 16] = c[r];
        }
#pragma unroll
        for (int kk = 0; kk < 16; ++kk) {
            b[kk] *= d256[kk];   // advance B by 256 timesteps
        }
    }
}

extern "C" void kernel_launch(void* const* d_in, const int* in_sizes, int n_in,
                              void* d_out, int out_size, void* d_ws, size_t ws_size,
                              hipStream_t stream) {
    // Inputs (setup_inputs order): d_in[0]=L (scalar, ==4096), d_in[1]=gamma,
    // d_in[2]=R, d_in[3]=p. All f32 (d,o) = (2048,64). Output f32 (1,2048,4096).
    (void)in_sizes; (void)n_in; (void)out_size; (void)d_ws; (void)ws_size;
    const float* gamma = (const float*)d_in[1];
    const float* R     = (const float*)d_in[2];
    const float* p     = (const float*)d_in[3];
    float* out = (float*)d_out;

    dim3 block(WAVES_PER_BLOCK * 32);
    dim3 grid(D_MODEL / WAVES_PER_BLOCK);
    ImplicitModalFilter_wmma_kernel<<<grid, block, 0, stream>>>(gamma, R, p, out);
}